// CGProduct_69561290326722
// MI455X (gfx1250) — compile-verified
//
#include <hip/hip_runtime.h>
#include <hip/hip_bf16.h>

// ---------------------------------------------------------------------------
// CG tensor product, MAXL=3, N=6144, TAU=32.
// Bandwidth-bound (~300 MB traffic => ~13 us floor @ 23.3 TB/s), so f32 WMMA
// (V_WMMA_F32_16X16X4_F32) is used to keep reference precision; flops are not
// the bottleneck. Per wave: 8 items x {re,im} = 16 columns = one 16x16 D tile;
// per (l1,l2) pair: D[Kx16] = CG[KxM] @ kron[Mx16] over ceil(M/4) k-chunks.
//  - CG staged to LDS in zero-padded WMMA layout -> unconditional ds_load_b64
//    for the A operand (no exec-mask branches)
//  - kron (B) computed in registers, 2 VALU per element after folding z into
//    pre-swapped/sign-folded operands
//  - inputs staged global->LDS with gfx1250 async-to-LDS ops (ASYNCcnt)
// ---------------------------------------------------------------------------

#define MAXL 3
#define NPTS 6144
#define TAU  32
constexpr int NITEMS = NPTS * TAU;   // 196608

typedef __attribute__((ext_vector_type(2))) float v2f;
typedef __attribute__((ext_vector_type(8))) float v8f;

struct CGPtrs { const float* p[16]; };

constexpr int dimM(int l1, int l2) { return (2 * l1 + 1) * (2 * l2 + 1); }
constexpr int lminF(int l1, int l2) { return l1 > l2 ? l1 - l2 : l2 - l1; }
constexpr int lmaxF(int l1, int l2) { int s = l1 + l2; return s > MAXL ? MAXL : s; }
constexpr bool contrib(int l1, int l2, int l) { return l >= lminF(l1, l2) && l <= lmaxF(l1, l2); }
constexpr int kRows(int l1, int l2) {
    int lm = lminF(l1, l2), lx = lmaxF(l1, l2);
    return (lx + 1) * (lx + 1) - lm * lm;
}
constexpr int blkIdx(int l1, int l2, int l) {
    int b = 0;
    for (int p = 0; p < l1 * 4 + l2; p++) if (contrib(p >> 2, p & 3, l)) b++;
    return b;
}
constexpr int chOut(int l) {
    int c = 0;
    for (int p = 0; p < 16; p++) if (contrib(p >> 2, p & 3, l)) c++;
    return c * TAU;
}
constexpr int outOff(int l) {
    int o = 0;
    for (int q = 0; q < l; q++) o += NPTS * chOut(q) * (2 * q + 1) * 2;
    return o;
}
// padded CG layout: per pair, 16 rows x MP cols (MP = 4*ceil(M/4)), zero-filled
constexpr int nchF(int p) { return (dimM(p >> 2, p & 3) + 3) / 4; }
constexpr int mpadF(int p) { return 4 * nchF(p); }
constexpr int padOff(int p) {
    int o = 0;
    for (int q = 0; q < p; q++) o += 16 * mpadF(q);
    return o;
}
constexpr int CG_PAD_TOTAL = padOff(16);          // 16 * 288 = 4608 floats
constexpr int aOff(int l) {
    int o = 0;
    for (int q = 0; q < l; q++) o += (2 * q + 1) * 2;
    return o;
}
static_assert(CG_PAD_TOTAL == 4608, "cg pad size");
static_assert(outOff(3) + NPTS * chOut(3) * 7 * 2 == 61341696, "out size");

// ---------------------------------------------------------------------------
// One (l1,l2) pair. WMMA f32 16x16x4 lane layouts (ISA 7.12.2):
//   A (16x4): lane<16 -> rows r=lane, cols k,k+1 ; lane>=16 -> rows r=lane-16,
//   cols k+2,k+3.  B assumed to mirror (4x16, N=lane%16, k split by lane half).
//   D: VGPR v -> row v + 8*(lane>=16), N = lane%16.
// ab[] holds pre-transformed operands so kron value = ar*bs1 + ai2*bs2.
// ---------------------------------------------------------------------------
template <int L1, int L2>
__device__ __forceinline__ void doPair(const float (&ab)[64], const float* cgLds,
                                       float* __restrict__ out,
                                       int lane, int half, int z, int nIdx, int cIdx) {
    constexpr int M    = dimM(L1, L2);
    constexpr int P    = L1 * 4 + L2;
    constexpr int MP   = mpadF(P);
    constexpr int POFF = padOff(P);
    constexpr int LMIN = lminF(L1, L2), LMAX = lmaxF(L1, L2);
    constexpr int NCH  = (M + 3) / 4;
    constexpr int D2   = 2 * L2 + 1;
    constexpr int AO   = aOff(L1);
    constexpr int BO   = 32 + aOff(L2);
    const int row = lane & 15;

    v8f acc = {0.f, 0.f, 0.f, 0.f, 0.f, 0.f, 0.f, 0.f};
#pragma unroll
    for (int kc = 0; kc < NCH; ++kc) {
        float kv[4];
#pragma unroll
        for (int j = 0; j < 4; ++j) {
            const int m = kc * 4 + j;                 // compile-time
            if (m < M) {
                const int m1 = m / D2, m2 = m % D2;   // compile-time
                kv[j] = ab[AO + 2 * m1] * ab[BO + 2 * m2]       // ar*bs1
                      + ab[AO + 2 * m1 + 1] * ab[BO + 2 * m2 + 1]; // ai2*bs2
            } else {
                kv[j] = 0.f;
            }
        }
        v2f B;
        B.x = half ? kv[2] : kv[0];
        B.y = half ? kv[3] : kv[1];

        // unconditional 8B-aligned LDS load from padded CG table
        const v2f A = *(const v2f*)(cgLds + POFF + row * MP + kc * 4 + 2 * half);

        acc = __builtin_amdgcn_wmma_f32_16x16x4_f32(
            /*neg_a=*/false, A, /*neg_b=*/false, B,
            /*c_mod=*/(short)0, acc, /*reuse_a=*/false, /*reuse_b=*/false);
    }

    // scatter rows r=0..K-1 -> (l_out, m) output segments
    int rbase = 0;
#pragma unroll
    for (int lo = LMIN; lo <= LMAX; ++lo) {
        const int blk  = blkIdx(L1, L2, lo);
        const int base = outOff(lo) +
                         ((nIdx * chOut(lo) + blk * TAU + cIdx) * (2 * lo + 1)) * 2 + z;
#pragma unroll
        for (int mm = 0; mm <= 2 * lo; ++mm) {
            const int r = rbase + mm;                 // compile-time
            if ((r >> 3) == half) out[base + mm * 2] = acc[r & 7];
        }
        rbase += 2 * lo + 1;
    }
}

__launch_bounds__(256)
__global__ void CGProduct_69561290326722_kernel(
    const float* __restrict__ r10, const float* __restrict__ r11,
    const float* __restrict__ r12, const float* __restrict__ r13,
    const float* __restrict__ r20, const float* __restrict__ r21,
    const float* __restrict__ r22, const float* __restrict__ r23,
    CGPtrs cgp, float* __restrict__ out) {
    __shared__ __align__(16) float cgLds[CG_PAD_TOTAL];      // 18 KB, padded
    __shared__ __align__(16) float itemLds[8 * 8 * 64];      // 16 KB

    const int tid  = threadIdx.x;
    const int lane = tid & 31;
    const int wv   = tid >> 5;

    // ---- stage CG matrices into zero-padded WMMA-ready LDS layout ----
#pragma unroll
    for (int p = 0; p < 16; ++p) {
        const int M = dimM(p >> 2, p & 3);
        const int K = kRows(p >> 2, p & 3);
        const int MP = mpadF(p), base = padOff(p);
        const float* src = cgp.p[p];
        for (int i = tid; i < 16 * MP; i += 256) {
            const int r = i / MP, m = i % MP;        // div by compile-time MP
            cgLds[base + i] = (r < K && m < M) ? src[r * M + m] : 0.f;
        }
    }

    // ---- stage this wave's 8 items via gfx1250 async global->LDS copies ----
    const int giW = (blockIdx.x * 8 + wv) * 8;
    float* wbase = &itemLds[wv * 8 * 64];
    const float* reps[8] = {r10, r11, r12, r13, r20, r21, r22, r23};
#pragma unroll
    for (int l = 0; l < 4; ++l) {
        const int S = (2 * l + 1) * 2;
        const float* g1 = reps[l]     + (size_t)giW * S;
        const float* g2 = reps[4 + l] + (size_t)giW * S;
        for (int idx = lane; idx < 8 * S; idx += 32) {
            const int it = idx / S, r = idx % S;     // div by compile-time S
            // generic LDS pointer low 32 bits == LDS byte offset (ISA aperture)
            const unsigned d1 = (unsigned)(uintptr_t)&wbase[it * 64 + aOff(l) + r];
            asm volatile("global_load_async_to_lds_b32 %0, %1, off"
                         :: "v"(d1),
                            "v"((unsigned long long)(uintptr_t)(g1 + idx))
                         : "memory");
            const unsigned d2 = (unsigned)(uintptr_t)&wbase[it * 64 + 32 + aOff(l) + r];
            asm volatile("global_load_async_to_lds_b32 %0, %1, off"
                         :: "v"(d2),
                            "v"((unsigned long long)(uintptr_t)(g2 + idx))
                         : "memory");
        }
    }
    asm volatile("s_wait_asynccnt 0x0" ::: "memory");
    __syncthreads();

    // ---- per-lane item registers (4-lane replication within the wave) ----
    const int half  = lane >> 4;
    const int col   = lane & 15;     // WMMA D column
    const int itemL = col >> 1;
    const int z     = col & 1;       // 0 = real, 1 = imag
    float ab[64];
    {
        const float4* src = (const float4*)&wbase[itemL * 64];
#pragma unroll
        for (int q = 0; q < 16; ++q) {
            const float4 t = src[q];
            ab[4 * q] = t.x; ab[4 * q + 1] = t.y; ab[4 * q + 2] = t.z; ab[4 * q + 3] = t.w;
        }
    }
    // fold z into operands: kron value becomes ar*bs1 + ai2*bs2 (mul+fma)
    //   z=0: bs1=br, bs2=bi, ai2=-ai  -> ar*br - ai*bi   (real part)
    //   z=1: bs1=bi, bs2=br, ai2=+ai  -> ar*bi + ai*br   (imag part)
#pragma unroll
    for (int q = 0; q < 16; ++q) {
        const float br = ab[32 + 2 * q], bi = ab[32 + 2 * q + 1];
        ab[32 + 2 * q]     = z ? bi : br;
        ab[32 + 2 * q + 1] = z ? br : bi;
        ab[2 * q + 1]      = z ? ab[2 * q + 1] : -ab[2 * q + 1];
    }
    const int gi   = giW + itemL;
    const int nIdx = gi / TAU;
    const int cIdx = gi % TAU;

    doPair<0, 0>(ab, cgLds, out, lane, half, z, nIdx, cIdx);
    doPair<0, 1>(ab, cgLds, out, lane, half, z, nIdx, cIdx);
    doPair<0, 2>(ab, cgLds, out, lane, half, z, nIdx, cIdx);
    doPair<0, 3>(ab, cgLds, out, lane, half, z, nIdx, cIdx);
    doPair<1, 0>(ab, cgLds, out, lane, half, z, nIdx, cIdx);
    doPair<1, 1>(ab, cgLds, out, lane, half, z, nIdx, cIdx);
    doPair<1, 2>(ab, cgLds, out, lane, half, z, nIdx, cIdx);
    doPair<1, 3>(ab, cgLds, out, lane, half, z, nIdx, cIdx);
    doPair<2, 0>(ab, cgLds, out, lane, half, z, nIdx, cIdx);
    doPair<2, 1>(ab, cgLds, out, lane, half, z, nIdx, cIdx);
    doPair<2, 2>(ab, cgLds, out, lane, half, z, nIdx, cIdx);
    doPair<2, 3>(ab, cgLds, out, lane, half, z, nIdx, cIdx);
    doPair<3, 0>(ab, cgLds, out, lane, half, z, nIdx, cIdx);
    doPair<3, 1>(ab, cgLds, out, lane, half, z, nIdx, cIdx);
    doPair<3, 2>(ab, cgLds, out, lane, half, z, nIdx, cIdx);
    doPair<3, 3>(ab, cgLds, out, lane, half, z, nIdx, cIdx);
}

extern "C" void kernel_launch(void* const* d_in, const int* in_sizes, int n_in,
                              void* d_out, int out_size, void* d_ws, size_t ws_size,
                              hipStream_t stream) {
    (void)in_sizes; (void)n_in; (void)out_size; (void)d_ws; (void)ws_size;
    const float* r10 = (const float*)d_in[0];
    const float* r11 = (const float*)d_in[1];
    const float* r12 = (const float*)d_in[2];
    const float* r13 = (const float*)d_in[3];
    const float* r20 = (const float*)d_in[4];
    const float* r21 = (const float*)d_in[5];
    const float* r22 = (const float*)d_in[6];
    const float* r23 = (const float*)d_in[7];
    CGPtrs cgp;
    for (int p = 0; p < 16; ++p) cgp.p[p] = (const float*)d_in[8 + p];
    float* out = (float*)d_out;

    // 196608 items / (8 waves * 8 items) = 3072 blocks, exact (no tail)
    const int blocks = NITEMS / 64;
    CGProduct_69561290326722_kernel<<<blocks, 256, 0, stream>>>(
        r10, r11, r12, r13, r20, r21, r22, r23, cgp, out);
}